// EnhancedHeteroGNN_7507602833969
// MI455X (gfx1250) — compile-verified
//
#include <hip/hip_runtime.h>
#include <math.h>

// ---------------------------------------------------------------------------
// EnhancedHeteroGNN forward, MI455X (gfx1250, wave32).
// Live path only (author/SAGE branch never feeds the returned logits).
// GEMMs on V_WMMA_F32_16X16X4_F32 (exact f32). B-slab staged via the Tensor
// Data Mover using inline asm (portable across both gfx1250 toolchains);
// scatter softmax via native f32 global atomics.
// ---------------------------------------------------------------------------

typedef float v2f __attribute__((ext_vector_type(2)));
typedef float v8f __attribute__((ext_vector_type(8)));
typedef unsigned u32x4 __attribute__((ext_vector_type(4)));
typedef unsigned u32x8 __attribute__((ext_vector_type(8)));

#define HID   128
#define FPAT  256
#define NHEAD 4
#define CH    32          // channels per head
#define MT    128         // block M tile (8 waves x 16 rows)
#define KSLAB 32
#define ASTR  34          // padded LDS stride for A (bank-conflict free, 8B aligned)

#if defined(__gfx1250__)
#define USE_TDM_ASM 1
#else
#define USE_TDM_ASM 0
#endif

// ---------------- float atomic max (CAS; emax init to -inf, finite vals) ----
__device__ inline void atomicMaxF32(float* addr, float val) {
  unsigned* au = (unsigned*)addr;
  unsigned old = *au;
  while (__uint_as_float(old) < val) {
    unsigned assumed = old;
    old = atomicCAS(au, assumed, __float_as_uint(val));
    if (old == assumed) break;
  }
}

// ---------------- per-row mean / rsqrt(var) for fused LayerNorm -------------
__global__ __launch_bounds__(256)
void row_stats_k(const float* __restrict__ x, float* __restrict__ mu,
                 float* __restrict__ rs, int M, int F) {
  int lane = threadIdx.x & 31;
  int n = blockIdx.x * 8 + (threadIdx.x >> 5);
  if (n >= M) return;
  const float* r = x + (size_t)n * F;
  float s = 0.f, s2 = 0.f;
  for (int j = lane; j < F; j += 32) { float v = r[j]; s += v; s2 += v * v; }
  #pragma unroll
  for (int m = 1; m < 32; m <<= 1) { s += __shfl_xor(s, m, 32); s2 += __shfl_xor(s2, m, 32); }
  if (lane == 0) {
    float mean = s / (float)F;
    float var  = s2 / (float)F - mean * mean;
    mu[n] = mean;
    rs[n] = rsqrtf(var + 1e-5f);
  }
}

// ---------------- generic f32 WMMA GEMM: C[M x N] = op(A)[M x K] @ B + bias -
template <int K, int N, bool LN_IN, bool RELU>
__global__ __launch_bounds__(256)
void gemm_wmma_k(const float* __restrict__ A,
                 const float* __restrict__ mu, const float* __restrict__ rs,
                 const float* __restrict__ lnw, const float* __restrict__ lnb,
                 const float* __restrict__ B, const float* __restrict__ bias,
                 float* __restrict__ C, int M) {
  __shared__ float Abuf[MT * ASTR];
  __shared__ float Bbuf[KSLAB * N];

  const int t    = threadIdx.x;
  const int wave = t >> 5;
  const int lane = t & 31;
  const int r0   = blockIdx.x * MT;

  constexpr int NT   = N / 16;
  constexpr int BTOT = KSLAB * N;     // floats per B slab (contiguous in memory)

  v8f acc[NT];
  #pragma unroll
  for (int i = 0; i < NT; ++i)
    #pragma unroll
    for (int j = 0; j < 8; ++j) acc[i][j] = 0.f;

  const int mrow = lane & 15;         // A fragment: M = lane%16
  const int kq   = (lane >> 4) * 2;   // A/B fragment: K sub-offset 0 or 2

  for (int k0 = 0; k0 < K; k0 += KSLAB) {
#if USE_TDM_ASM
    // ---- stage B slab via Tensor Data Mover (1-D contiguous 32xN tile) -----
    // D# per CDNA5 ISA; groups 2/3 NULL (<=2D).
    if (wave == 0) {
      unsigned ldsa = (unsigned)(size_t)(&Bbuf[0]);                 // LDS byte addr
      unsigned long long ga = (unsigned long long)(size_t)(B + (size_t)k0 * N);
      u32x4 g0;
      g0[0] = 1u;                                                   // count=1
      g0[1] = ldsa;                                                 // lds_addr
      g0[2] = (unsigned)(ga & 0xffffffffu);                         // gaddr[31:0]
      g0[3] = (unsigned)((ga >> 32) & 0x1ffffffu) | (2u << 30);     // gaddr[56:32]|type=2
      u32x8 g1;
      g1[0] = (2u << 16);                                           // data_size=4B, mask=0
      g1[1] = (unsigned)((BTOT & 0xffff) << 16);                    // tensor_dim0[15:0]
      g1[2] = (unsigned)(((BTOT >> 16) & 0xffff) | (1u << 16));     // td0[31:16] | td1=1
      g1[3] = (unsigned)((BTOT & 0xffff) << 16);                    // tile_dim0 = BTOT
      g1[4] = 0u;                                                   // tile_dim1/2 = 0
      g1[5] = (unsigned)BTOT;                                       // dim0_stride[31:0]
      g1[6] = 0u;
      g1[7] = 0u;
      asm volatile("tensor_load_to_lds %0, %1"
                   :
                   : "s"(g0), "s"(g1)
                   : "memory");
    }
#endif
    // ---- stage A slab (128 rows x 32 k), LN fused, row-masked --------------
    {
      int rg = t >> 3;       // 0..31
      int c4 = t & 7;        // float4 within the 32-wide slab
      #pragma unroll
      for (int p = 0; p < 4; ++p) {
        int r  = rg + p * 32;
        int gr = r0 + r;
        float4 v = make_float4(0.f, 0.f, 0.f, 0.f);
        if (gr < M) {
          v = *(const float4*)(A + (size_t)gr * K + k0 + c4 * 4);
          if (LN_IN) {
            float m = mu[gr], s = rs[gr];
            int kk = k0 + c4 * 4;
            v.x = (v.x - m) * s * lnw[kk + 0] + lnb[kk + 0];
            v.y = (v.y - m) * s * lnw[kk + 1] + lnb[kk + 1];
            v.z = (v.z - m) * s * lnw[kk + 2] + lnb[kk + 2];
            v.w = (v.w - m) * s * lnw[kk + 3] + lnb[kk + 3];
          }
        }
        float* d = &Abuf[r * ASTR + c4 * 4];
        d[0] = v.x; d[1] = v.y; d[2] = v.z; d[3] = v.w;
      }
    }
#if USE_TDM_ASM
    if (wave == 0) {
      asm volatile("s_wait_tensorcnt 0x0" ::: "memory");
    }
#else
    // ---- stage B slab (32 k x N) manually ----------------------------------
    {
      constexpr int TOT4 = KSLAB * N / 4;
      for (int f = t; f < TOT4; f += 256) {
        int rr  = f / (N / 4);
        int cc4 = f % (N / 4);
        float4 v = *(const float4*)(B + (size_t)(k0 + rr) * N + cc4 * 4);
        float* d = &Bbuf[rr * N + cc4 * 4];
        d[0] = v.x; d[1] = v.y; d[2] = v.z; d[3] = v.w;
      }
    }
#endif
    __syncthreads();

    if (k0 + KSLAB < K)   // pull next A slab toward L2 (global_prefetch_b8)
      __builtin_prefetch(A + (size_t)(r0 + wave * 16 + mrow) * K + k0 + KSLAB, 0, 3);

    const float* abase = &Abuf[(wave * 16 + mrow) * ASTR + kq];
    #pragma unroll
    for (int kk = 0; kk < KSLAB; kk += 4) {
      v2f a = *(const v2f*)(abase + kk);             // K = kk+kq, kk+kq+1
      #pragma unroll
      for (int nt = 0; nt < NT; ++nt) {
        int ncol = nt * 16 + mrow;
        v2f b;
        b.x = Bbuf[(kk + kq + 0) * N + ncol];
        b.y = Bbuf[(kk + kq + 1) * N + ncol];
        acc[nt] = __builtin_amdgcn_wmma_f32_16x16x4_f32(
            false, a, false, b, (short)0, acc[nt], false, false);
      }
    }
    __syncthreads();
  }

  // ---- epilogue: C/D layout -> VGPR r: (M=r, lanes0-15) / (M=r+8, lanes16-31)
  int growb = r0 + wave * 16 + ((lane >> 4) ? 8 : 0);
  int ncl   = lane & 15;
  #pragma unroll
  for (int nt = 0; nt < NT; ++nt) {
    int gc = nt * 16 + ncl;
    float bv = bias ? bias[gc] : 0.f;
    #pragma unroll
    for (int r = 0; r < 8; ++r) {
      int gr = growb + r;
      if (gr < M) {
        float v = acc[nt][r] + bv;
        if (RELU) v = fmaxf(v, 0.f);
        C[(size_t)gr * N + gc] = v;
      }
    }
  }
}

// ---------------- attention logits es/ed per node, per head ----------------
__global__ __launch_bounds__(256)
void node_attn_k(const float* __restrict__ h, const float* __restrict__ a_s,
                 const float* __restrict__ a_d, float* __restrict__ es,
                 float* __restrict__ ed, int M) {
  int lane = threadIdx.x & 31;
  int n = blockIdx.x * 8 + (threadIdx.x >> 5);
  if (n >= M) return;
  int c0 = lane * 4;
  int head = c0 >> 5;
  int ci = c0 & 31;
  float4 hv  = *(const float4*)(h + (size_t)n * HID + c0);
  float4 as4 = *(const float4*)(a_s + head * CH + ci);
  float4 ad4 = *(const float4*)(a_d + head * CH + ci);
  float ps = hv.x * as4.x + hv.y * as4.y + hv.z * as4.z + hv.w * as4.w;
  float pd = hv.x * ad4.x + hv.y * ad4.y + hv.z * ad4.z + hv.w * ad4.w;
  #pragma unroll
  for (int m = 1; m < 8; m <<= 1) { ps += __shfl_xor(ps, m, 32); pd += __shfl_xor(pd, m, 32); }
  if ((lane & 7) == 0) { es[n * NHEAD + head] = ps; ed[n * NHEAD + head] = pd; }
}

// ---------------- init emax/-inf, den/agg = 0 --------------------------------
__global__ __launch_bounds__(256)
void gat_init_k(float* __restrict__ emax, float* __restrict__ den,
                float* __restrict__ agg, int M) {
  int i = blockIdx.x * 256 + threadIdx.x;
  if (i < M * NHEAD) { emax[i] = -INFINITY; den[i] = 0.f; }
  if (i < M * HID) agg[i] = 0.f;
}

__device__ inline void edge_sd(int e, int EC, const int* src, const int* dst,
                               int& s, int& d) {
  if (e < EC) { s = src[e]; d = dst[e]; } else { s = e - EC; d = s; }  // self-loop
}

__global__ __launch_bounds__(256)
void edge_max_k(const int* __restrict__ src, const int* __restrict__ dst,
                const float* __restrict__ es, const float* __restrict__ ed,
                float* __restrict__ emax, int EC, int M) {
  int idx = blockIdx.x * 256 + threadIdx.x;
  int tot = (EC + M) * NHEAD;
  if (idx >= tot) return;
  int e = idx >> 2, hh = idx & 3, s, d;
  edge_sd(e, EC, src, dst, s, d);
  float v = es[s * NHEAD + hh] + ed[d * NHEAD + hh];
  v = v > 0.f ? v : 0.2f * v;
  atomicMaxF32(&emax[d * NHEAD + hh], v);
}

__global__ __launch_bounds__(256)
void edge_sum_k(const int* __restrict__ src, const int* __restrict__ dst,
                const float* __restrict__ es, const float* __restrict__ ed,
                const float* __restrict__ emax, float* __restrict__ den,
                int EC, int M) {
  int idx = blockIdx.x * 256 + threadIdx.x;
  int tot = (EC + M) * NHEAD;
  if (idx >= tot) return;
  int e = idx >> 2, hh = idx & 3, s, d;
  edge_sd(e, EC, src, dst, s, d);
  float v = es[s * NHEAD + hh] + ed[d * NHEAD + hh];
  v = v > 0.f ? v : 0.2f * v;
  atomicAdd(&den[d * NHEAD + hh], __expf(v - emax[d * NHEAD + hh]));
}

// one wave per edge: 32 lanes x 4 channels scatter-add of h[s]*alpha into agg[d]
__global__ __launch_bounds__(256)
void edge_aggr_k(const int* __restrict__ src, const int* __restrict__ dst,
                 const float* __restrict__ es, const float* __restrict__ ed,
                 const float* __restrict__ emax, const float* __restrict__ den,
                 const float* __restrict__ h, float* __restrict__ agg,
                 int EC, int M) {
  int lane = threadIdx.x & 31;
  int e = blockIdx.x * 8 + (threadIdx.x >> 5);
  if (e >= EC + M) return;
  int s, d;
  edge_sd(e, EC, src, dst, s, d);
  int c0 = lane * 4;
  int hh = c0 >> 5;
  float v = es[s * NHEAD + hh] + ed[d * NHEAD + hh];
  v = v > 0.f ? v : 0.2f * v;
  float alpha = __expf(v - emax[d * NHEAD + hh]) / (den[d * NHEAD + hh] + 1e-16f);
  float4 hv = *(const float4*)(h + (size_t)s * HID + c0);
  float* ap = agg + (size_t)d * HID + c0;
  atomicAdd(ap + 0, hv.x * alpha);
  atomicAdd(ap + 1, hv.y * alpha);
  atomicAdd(ap + 2, hv.z * alpha);
  atomicAdd(ap + 3, hv.w * alpha);
}

// ---------------- xp = LN(relu(agg + gb)) * nw + nb + xp  (in place) --------
__global__ __launch_bounds__(256)
void gat_epilogue_k(const float* __restrict__ agg, const float* __restrict__ gb,
                    const float* __restrict__ nw, const float* __restrict__ nb,
                    float* __restrict__ xp, int M) {
  int lane = threadIdx.x & 31;
  int n = blockIdx.x * 8 + (threadIdx.x >> 5);
  if (n >= M) return;
  int c0 = lane * 4;
  float4 v = *(const float4*)(agg + (size_t)n * HID + c0);
  v.x = fmaxf(v.x + gb[c0 + 0], 0.f);
  v.y = fmaxf(v.y + gb[c0 + 1], 0.f);
  v.z = fmaxf(v.z + gb[c0 + 2], 0.f);
  v.w = fmaxf(v.w + gb[c0 + 3], 0.f);
  float s = v.x + v.y + v.z + v.w;
  #pragma unroll
  for (int m = 1; m < 32; m <<= 1) s += __shfl_xor(s, m, 32);
  float mean = s * (1.f / HID);
  float dx = v.x - mean, dy = v.y - mean, dz = v.z - mean, dw = v.w - mean;
  float q = dx * dx + dy * dy + dz * dz + dw * dw;
  #pragma unroll
  for (int m = 1; m < 32; m <<= 1) q += __shfl_xor(q, m, 32);
  float rstd = rsqrtf(q * (1.f / HID) + 1e-5f);
  float4 xo = *(const float4*)(xp + (size_t)n * HID + c0);
  float4 out;
  out.x = dx * rstd * nw[c0 + 0] + nb[c0 + 0] + xo.x;
  out.y = dy * rstd * nw[c0 + 1] + nb[c0 + 1] + xo.y;
  out.z = dz * rstd * nw[c0 + 2] + nb[c0 + 2] + xo.z;
  out.w = dw * rstd * nw[c0 + 3] + nb[c0 + 3] + xo.w;
  *(float4*)(xp + (size_t)n * HID + c0) = out;
}

// ---------------- tiny head: out[Mx8] = hid[Mx64] @ c2_W + c2_b -------------
__global__ __launch_bounds__(256)
void classifier2_k(const float* __restrict__ hid, const float* __restrict__ W,
                   const float* __restrict__ b, float* __restrict__ out, int M) {
  int idx = blockIdx.x * 256 + threadIdx.x;
  if (idx >= M * 8) return;
  int n = idx >> 3, j = idx & 7;
  const float* hr = hid + (size_t)n * 64;
  float acc = b[j];
  #pragma unroll
  for (int k = 0; k < 64; ++k) acc += hr[k] * W[k * 8 + j];
  out[idx] = acc;
}

// ---------------------------------------------------------------------------
extern "C" void kernel_launch(void* const* d_in, const int* in_sizes, int n_in,
                              void* d_out, int out_size, void* d_ws, size_t ws_size,
                              hipStream_t stream) {
  (void)n_in; (void)ws_size; (void)out_size;
  const int M  = in_sizes[0] / FPAT;     // num patents
  const int EC = in_sizes[2] / 2;        // num citation edges

  const float* x_patent = (const float*)d_in[0];
  const int*   ei       = (const int*)d_in[2];
  const int *src = ei, *dst = ei + EC;
  const float* pn_w = (const float*)d_in[4];
  const float* pn_b = (const float*)d_in[5];
  const float* pl_W = (const float*)d_in[8];
  const float* pl_b = (const float*)d_in[9];
  const float* g1_W = (const float*)d_in[12];
  const float* g1_as= (const float*)d_in[13];
  const float* g1_ad= (const float*)d_in[14];
  const float* g1_b = (const float*)d_in[15];
  const float* g2_W = (const float*)d_in[16];
  const float* g2_as= (const float*)d_in[17];
  const float* g2_ad= (const float*)d_in[18];
  const float* g2_b = (const float*)d_in[19];
  const float* n1_w = (const float*)d_in[23];
  const float* n1_b = (const float*)d_in[24];
  const float* n3_w = (const float*)d_in[27];
  const float* n3_b = (const float*)d_in[28];
  const float* c1_W = (const float*)d_in[29];
  const float* c1_b = (const float*)d_in[30];
  const float* c2_W = (const float*)d_in[31];
  const float* c2_b = (const float*)d_in[32];

  // workspace partition (floats)
  float* w = (float*)d_ws;
  float* xp   = w; w += (size_t)M * HID;
  float* h    = w; w += (size_t)M * HID;
  float* agg  = w; w += (size_t)M * HID;
  float* es   = w; w += (size_t)M * NHEAD;
  float* ed   = w; w += (size_t)M * NHEAD;
  float* emax = w; w += (size_t)M * NHEAD;
  float* den  = w; w += (size_t)M * NHEAD;
  float* mu   = w; w += (size_t)M;
  float* rs   = w; w += (size_t)M;

  const dim3 blk(256);
  const int gNode = (M + 7) / 8;                 // wave-per-node kernels
  const int gGemm = (M + MT - 1) / MT;
  const int E     = EC + M;                      // edges incl. self loops
  const int gEdgeF= (E * NHEAD + 255) / 256;     // thread-per-(edge,head)
  const int gEdgeW= (E + 7) / 8;                 // wave-per-edge
  const int gInit = (M * HID + 255) / 256;

  // 1) xp = LN(x_patent) @ pl_W + pl_b   (LN fused into A staging)
  row_stats_k<<<gNode, blk, 0, stream>>>(x_patent, mu, rs, M, FPAT);
  gemm_wmma_k<FPAT, HID, true, false><<<gGemm, blk, 0, stream>>>(
      x_patent, mu, rs, pn_w, pn_b, pl_W, pl_b, xp, M);

  // 2) GAT layer 1
  gemm_wmma_k<HID, HID, false, false><<<gGemm, blk, 0, stream>>>(
      xp, nullptr, nullptr, nullptr, nullptr, g1_W, nullptr, h, M);
  node_attn_k<<<gNode, blk, 0, stream>>>(h, g1_as, g1_ad, es, ed, M);
  gat_init_k<<<gInit, blk, 0, stream>>>(emax, den, agg, M);
  edge_max_k<<<gEdgeF, blk, 0, stream>>>(src, dst, es, ed, emax, EC, M);
  edge_sum_k<<<gEdgeF, blk, 0, stream>>>(src, dst, es, ed, emax, den, EC, M);
  edge_aggr_k<<<gEdgeW, blk, 0, stream>>>(src, dst, es, ed, emax, den, h, agg, EC, M);
  gat_epilogue_k<<<gNode, blk, 0, stream>>>(agg, g1_b, n1_w, n1_b, xp, M);

  // 3) GAT layer 2
  gemm_wmma_k<HID, HID, false, false><<<gGemm, blk, 0, stream>>>(
      xp, nullptr, nullptr, nullptr, nullptr, g2_W, nullptr, h, M);
  node_attn_k<<<gNode, blk, 0, stream>>>(h, g2_as, g2_ad, es, ed, M);
  gat_init_k<<<gInit, blk, 0, stream>>>(emax, den, agg, M);
  edge_max_k<<<gEdgeF, blk, 0, stream>>>(src, dst, es, ed, emax, EC, M);
  edge_sum_k<<<gEdgeF, blk, 0, stream>>>(src, dst, es, ed, emax, den, EC, M);
  edge_aggr_k<<<gEdgeW, blk, 0, stream>>>(src, dst, es, ed, emax, den, h, agg, EC, M);
  gat_epilogue_k<<<gNode, blk, 0, stream>>>(agg, g2_b, n3_w, n3_b, xp, M);

  // 4) classifier: relu(xp @ c1_W + c1_b) @ c2_W + c2_b
  gemm_wmma_k<HID, 64, false, true><<<gGemm, blk, 0, stream>>>(
      xp, nullptr, nullptr, nullptr, nullptr, c1_W, c1_b, h, M);
  classifier2_k<<<(M * 8 + 255) / 256, blk, 0, stream>>>(h, c2_W, c2_b, (float*)d_out, M);
}